// SupervisedGraphSage_Rand_49022756716630
// MI455X (gfx1250) — compile-verified
//
#include <hip/hip_runtime.h>
#include <hip/hip_bf16.h>

typedef __attribute__((ext_vector_type(2))) float v2f;
typedef __attribute__((ext_vector_type(4))) float v4f;
typedef __attribute__((ext_vector_type(8))) float v8f;

constexpr int kF    = 602;   // feature dim (301 float2 pairs per row, rows 8B-aligned)
constexpr int kFPad = 604;   // K padded to multiple of 4 (WMMA K-step)
constexpr int kFP   = 606;   // LDS A-tile row stride (606 % 64 = 30 -> conflict-free column reads)
constexpr int kH1   = 128;
constexpr int kH2   = 128;
constexpr int kC    = 41;
constexpr int kCPad = 48;    // classifier cols padded to 3 wave-tiles
constexpr int kS    = 3;

// Pair-interleaved transposed weight layouts: one b64 per B-fragment lane,
// 16 lanes -> 128 contiguous bytes.
//   W1P [302][128][2] : W1P[p*256 + 2n + j] = (2p+j < 602) ? W1[n*602 + 2p+j] : 0
//   W2P [ 64][128][2] : W2P[p*256 + 2n + j] = W2[n*128 + 2p+j]
//   WcP [ 64][ 48][2] : WcP[p* 96 + 2n + j] = (n < 41) ? Wc[n*128 + 2p+j] : 0
constexpr int kW1PElems = (kFPad / 2) * 2 * kH1;   // 302*256 = 77312
constexpr int kW2PElems = (kH1 / 2) * 2 * kH1;     //  64*256 = 16384
constexpr int kWcPElems = (kH2 / 2) * 2 * kCPad;   //  64*96  =  6144

__global__ __launch_bounds__(256) void prep_w1p_kernel(const float* __restrict__ W1,
                                                       float* __restrict__ W1P) {
    int i = blockIdx.x * 256 + threadIdx.x;
    if (i >= kW1PElems) return;
    int p = i >> 8, idx = i & 255, n = idx >> 1, j = idx & 1, k = 2 * p + j;
    W1P[i] = (k < kF) ? W1[(long)n * kF + k] : 0.0f;
}

__global__ __launch_bounds__(256) void prep_w2p_kernel(const float* __restrict__ W2,
                                                       float* __restrict__ W2P) {
    int i = blockIdx.x * 256 + threadIdx.x;   // exactly 16384
    int p = i >> 8, idx = i & 255, n = idx >> 1, j = idx & 1, k = 2 * p + j;
    W2P[i] = W2[(long)n * kH1 + k];
}

__global__ __launch_bounds__(256) void prep_wcp_kernel(const float* __restrict__ Wc,
                                                       float* __restrict__ WcP) {
    int i = blockIdx.x * 256 + threadIdx.x;   // exactly 6144
    int p = i / 96, idx = i % 96, n = idx >> 1, j = idx & 1, k = 2 * p + j;
    WcP[i] = (n < kC) ? Wc[(long)n * kH2 + k] : 0.0f;
}

// ---------------------------------------------------------------------------
// Kernel 1: agg1 = (feat[self] + sum_{s<3} feat[neigh1[s]]) / 4 ; h1 = relu(agg1 @ W1^T)
// One block = 16 nodes. 256 threads = 8 wave32; wave w -> output cols [16w,16w+16).
// ---------------------------------------------------------------------------
__global__ __launch_bounds__(256) void sage_layer1_kernel(
    const float* __restrict__ feat, const float* __restrict__ W1P,
    const int* __restrict__ neigh1, float* __restrict__ h1)
{
    __shared__ float Atile[16 * kFP];

    const int t    = threadIdx.x;
    const int row0 = blockIdx.x * 16;

    // zero K-padding columns 602..605
    if (t < 64) {
        int r = t >> 2, c = kF + (t & 3);
        Atile[r * kFP + c] = 0.0f;
    }

    // ---- aggregation: 16 threads per node, float2 loads (rows are 8B-aligned) ----
    {
        const int rr     = t >> 4;
        const int lane16 = t & 15;
        const int  node  = row0 + rr;
        const int* nb    = neigh1 + (long)node * kS;
        const int n0 = nb[0], n1 = nb[1], n2 = nb[2];
        const v2f* fs = (const v2f*)(feat + (long)node * kF);
        const v2f* f0 = (const v2f*)(feat + (long)n0   * kF);
        const v2f* f1 = (const v2f*)(feat + (long)n1   * kF);
        const v2f* f2 = (const v2f*)(feat + (long)n2   * kF);
        for (int c2 = lane16; c2 < kF / 2; c2 += 16) {          // 301 pairs
            v2f s = fs[c2], a = f0[c2], b = f1[c2], c = f2[c2];
            v2f res;
            res.x = (s.x + a.x + b.x + c.x) * 0.25f;
            res.y = (s.y + a.y + b.y + c.y) * 0.25f;
            *(v2f*)&Atile[rr * kFP + 2 * c2] = res;             // ds_store_b64
        }
    }
    __syncthreads();

    // ---- WMMA GEMM: [16 x 604] @ [604 x 16] per wave, guard-free ----
    const int wave = t >> 5;
    const int l    = t & 31;
    const int half = l >> 4;   // lanes 16-31 handle K+{2,3}
    const int mn   = l & 15;   // A row / D column for this lane
    const int col0 = wave * 16;

    v8f acc = {};
    const float* bbase = W1P + (long)half * 256 + 2 * (col0 + mn);
    const float* arow  = Atile + mn * kFP + half * 2;
    for (int k = 0; k < kFPad; k += 4) {
        v2f a = *(const v2f*)(arow + k);                        // ds_load_b64
        v2f b = *(const v2f*)(bbase + (long)(k >> 1) * 256);    // global_load_b64, coalesced
        acc = __builtin_amdgcn_wmma_f32_16x16x4_f32(
            false, a, false, b, (short)0, acc, false, false);
    }

    // store with ReLU; C/D layout: VGPR v -> row v (+8 for upper half-wave)
    float* out = h1 + (long)row0 * kH1 + col0 + mn;
#pragma unroll
    for (int v = 0; v < 8; ++v) {
        const float val = acc[v];
        out[(long)(v + half * 8) * kH1] = val > 0.0f ? val : 0.0f;
    }
}

// ---------------------------------------------------------------------------
// Kernel 2: agg2 over h1 (L2-resident), h2 = relu(agg2 @ W2^T),
//           scores = h2 @ Wc^T, row log_softmax.
// One block = 16 batch rows. 256 threads = 8 wave32.
// ---------------------------------------------------------------------------
__global__ __launch_bounds__(256) void sage_layer2_kernel(
    const float* __restrict__ h1, const float* __restrict__ W2P,
    const float* __restrict__ WcP, const int* __restrict__ neigh2,
    const int* __restrict__ nodes, float* __restrict__ outp)
{
    constexpr int LP = 132;   // 132 % 64 = 4 -> 16 distinct banks on column reads
    __shared__ float Ag[16 * LP];     // agg2 [16 x 128]
    __shared__ float Hh[16 * LP];     // h2   [16 x 128]
    __shared__ float Sc[16 * kCPad];  // scores [16 x 48]

    const int t    = threadIdx.x;
    const int row0 = blockIdx.x * 16;

    // ---- aggregation of h1, float4 loads (rows are 16B-aligned) ----
    {
        const int rr     = t >> 4;
        const int lane16 = t & 15;
        const int  node  = nodes[row0 + rr];
        const int* nb    = neigh2 + (long)node * kS;
        const int n0 = nb[0], n1 = nb[1], n2 = nb[2];
        const v4f* hs = (const v4f*)(h1 + (long)node * kH1);
        const v4f* h0 = (const v4f*)(h1 + (long)n0   * kH1);
        const v4f* ha = (const v4f*)(h1 + (long)n1   * kH1);
        const v4f* hb = (const v4f*)(h1 + (long)n2   * kH1);
#pragma unroll
        for (int c4 = lane16; c4 < kH1 / 4; c4 += 16) {         // 2 iterations
            v4f s = hs[c4], a = h0[c4], b = ha[c4], c = hb[c4];
            v4f res;
            res.x = (s.x + a.x + b.x + c.x) * 0.25f;
            res.y = (s.y + a.y + b.y + c.y) * 0.25f;
            res.z = (s.z + a.z + b.z + c.z) * 0.25f;
            res.w = (s.w + a.w + b.w + c.w) * 0.25f;
            *(v4f*)&Ag[rr * LP + 4 * c4] = res;                 // ds_store_b128
        }
    }
    __syncthreads();

    const int wave = t >> 5;
    const int l    = t & 31;
    const int half = l >> 4;
    const int mn   = l & 15;

    // ---- GEMM2: h2 = relu(agg2 @ W2^T), K = 128 ----
    {
        const int col0 = wave * 16;
        v8f acc = {};
        const float* bbase = W2P + (long)half * 256 + 2 * (col0 + mn);
        const float* arow  = Ag + mn * LP + half * 2;
        for (int k = 0; k < kH1; k += 4) {
            v2f a = *(const v2f*)(arow + k);
            v2f b = *(const v2f*)(bbase + (long)(k >> 1) * 256);
            acc = __builtin_amdgcn_wmma_f32_16x16x4_f32(
                false, a, false, b, (short)0, acc, false, false);
        }
#pragma unroll
        for (int v = 0; v < 8; ++v) {
            const float val = acc[v];
            Hh[(v + half * 8) * LP + col0 + mn] = val > 0.0f ? val : 0.0f;
        }
    }
    __syncthreads();

    // ---- GEMM3: scores = h2 @ Wc^T (48 padded cols over 3 waves, guard-free) ----
    if (wave < 3) {
        const int col0 = wave * 16;
        v8f acc = {};
        const float* bbase = WcP + (long)half * 96 + 2 * (col0 + mn);
        const float* arow  = Hh + mn * LP + half * 2;
        for (int k = 0; k < kH2; k += 4) {
            v2f a = *(const v2f*)(arow + k);
            v2f b = *(const v2f*)(bbase + (long)(k >> 1) * 96);
            acc = __builtin_amdgcn_wmma_f32_16x16x4_f32(
                false, a, false, b, (short)0, acc, false, false);
        }
#pragma unroll
        for (int v = 0; v < 8; ++v)
            Sc[(v + half * 8) * kCPad + col0 + mn] = acc[v];
    }
    __syncthreads();

    // ---- log_softmax per row over 41 classes ----
    if (t < 16) {
        const float* srow = Sc + t * kCPad;
        float mx = -3.402823466e38f;
        for (int c = 0; c < kC; ++c) mx = fmaxf(mx, srow[c]);
        float sum = 0.0f;
        for (int c = 0; c < kC; ++c) sum += __expf(srow[c] - mx);
        const float lg = __logf(sum);
        float* o = outp + (long)(row0 + t) * kC;
        for (int c = 0; c < kC; ++c) o[c] = srow[c] - mx - lg;
    }
}

extern "C" void kernel_launch(void* const* d_in, const int* in_sizes, int n_in,
                              void* d_out, int out_size, void* d_ws, size_t ws_size,
                              hipStream_t stream) {
    const float* feat   = (const float*)d_in[0];   // [N, 602]
    const float* W1     = (const float*)d_in[1];   // [128, 602]
    const float* W2     = (const float*)d_in[2];   // [128, 128]
    const float* Wc     = (const float*)d_in[3];   // [41, 128]
    const int*   neigh1 = (const int*)d_in[4];     // [N, 3]
    const int*   neigh2 = (const int*)d_in[5];     // [N, 3]
    const int*   nodes  = (const int*)d_in[6];     // [B]
    float*       out    = (float*)d_out;           // [B, 41]

    const int N  = in_sizes[0] / kF;               // 150000
    const int Bq = in_sizes[6];                    // 100000

    // workspace layout (all offsets 256B-aligned)
    char* ws = (char*)d_ws;
    float* h1  = (float*)ws;                                   // N*128 f32 = 76,800,000 B
    size_t off = (size_t)N * kH1 * sizeof(float);
    float* W1P = (float*)(ws + off);  off += (size_t)kW1PElems * sizeof(float);   // 309,248 B
    float* W2P = (float*)(ws + off);  off += (size_t)kW2PElems * sizeof(float);   //  65,536 B
    float* WcP = (float*)(ws + off);                                              //  24,576 B

    prep_w1p_kernel<<<(kW1PElems + 255) / 256, 256, 0, stream>>>(W1, W1P);
    prep_w2p_kernel<<<kW2PElems / 256,         256, 0, stream>>>(W2, W2P);
    prep_wcp_kernel<<<kWcPElems / 256,         256, 0, stream>>>(Wc, WcP);

    sage_layer1_kernel<<<N / 16, 256, 0, stream>>>(feat, W1P, neigh1, h1);
    sage_layer2_kernel<<<Bq / 16, 256, 0, stream>>>(h1, W2P, WcP, neigh2, nodes, out);
}